// ADMM_TV_9500467658911
// MI455X (gfx1250) — compile-verified
//
#include <hip/hip_runtime.h>
#include <math.h>

// ---------------- problem constants ----------------
#define NB   4
#define NPX  320
#define HW   (320L*320L)          // 102400
#define C2   (NB*2L*HW)           // 819200  (B,2,H,W)
#define C4   (NB*4L*HW)           // 1638400 (B,4,H,W)
#define NN2  (320L*320L)          // DFT matrix elements
#define RHO_ 1.0f
#define LAMBDA_TV_ 0.01f

// scalar slots
#define S_ACC0   0
#define S_ACC1   1
#define S_RSOLD  2
#define S_ALPHA  3
#define S_BETA   4
#define S_UPD    5
#define S_CGDONE 6
#define S_ADONE  7
#define S_COUNT  8
#define S_PRES   9
#define S_DRES   10

typedef float v2f __attribute__((ext_vector_type(2)));
typedef float v8f __attribute__((ext_vector_type(8)));

// ---------------- DFT matrix build ----------------
// Centered ortho DFT: F[k,n] = exp(-2*pi*i*(k-160)*(n-160)/320)/sqrt(320)
__global__ __launch_bounds__(256) void k_build_dft(float* __restrict__ Fr,
                                                   float* __restrict__ Fi) {
  long idx = (long)blockIdx.x * blockDim.x + threadIdx.x;
  if (idx >= NN2) return;
  int k = (int)(idx / NPX), n = (int)(idx % NPX);
  int t = ((k - 160) * (n - 160)) % NPX;   // keep phase argument small
  float ph = -6.283185307179586f * (float)t / (float)NPX;
  float s, c;
  sincosf(ph, &s, &c);
  const float inv = 0.05590169943749474f;  // 1/sqrt(320)
  Fr[idx] = c * inv;
  Fi[idx] = s * inv;
}

// ---------------- complex GEMM via V_WMMA_F32_16X16X4_F32 ----------------
// C = A * B  (complex, 320x320x320, row-major).  sgnA/sgnB conjugate A/B.
// Block = 4 waves = 64x64 macro-tile. Each wave owns a 16x64 strip (4 N-subtiles
// in registers, A operand reused across them). B panel (K=16 x N=64, re+im) is
// staged in LDS in a K-pair-interleaved layout so each inner-loop B operand is
// one ds_load_b64 matching the WMMA B register layout (K-pair picked by lane>=16).
#define KC 16
__global__ __launch_bounds__(128) void k_cgemm(
    const float* __restrict__ Ar, const float* __restrict__ Ai, long sA, float sgnA,
    const float* __restrict__ Br, const float* __restrict__ Bi, long sB, float sgnB,
    float* __restrict__ Cr, float* __restrict__ Ci, long sC) {
  const long bz = blockIdx.z;
  Ar += bz * sA; Ai += bz * sA;
  Br += bz * sB; Bi += bz * sB;
  Cr += bz * sC; Ci += bz * sC;

  __shared__ float lb[2 * KC * 64];             // [plane][kpair][col][2] = 8 KB

  const int lane = threadIdx.x;                 // 0..31 (wave32)
  const int wv   = threadIdx.y;                 // 0..3
  const int tid  = wv * 32 + lane;              // 0..127
  const int m0 = blockIdx.y * 64 + wv * 16;
  const int n0 = blockIdx.x * 64;
  const int rr = lane & 15;                     // row within A tile / col within C
  const int hh = lane >> 4;                     // K-pair select (ISA 7.12.2)

  v8f cr[4] = {}; v8f ci[4] = {};
  const float* arp = Ar + (long)(m0 + rr) * NPX;
  const float* aip = Ai + (long)(m0 + rr) * NPX;

  for (int kc = 0; kc < NPX; kc += KC) {
    __syncthreads();
    // ---- stage B panel: K rows [kc,kc+16), cols [n0,n0+64), both planes ----
    #pragma unroll
    for (int i = 0; i < 4; ++i) {
      int u = tid + i * 128;                    // 0..511 staging units
      int plane = u >> 8;                       // 0: real, 1: imag
      int rem = u & 255;
      int kp = rem >> 5;                        // k-pair 0..7
      int cp = rem & 31;                        // col-pair 0..31
      const float* Bp = plane ? Bi : Br;
      float sg = plane ? sgnB : 1.0f;
      long row = (long)(kc + 2 * kp);
      long col = (long)(n0 + cp * 2);
      v2f g0 = *(const v2f*)(Bp + row * NPX + col);          // B[k][c], B[k][c+1]
      v2f g1 = *(const v2f*)(Bp + (row + 1) * NPX + col);    // B[k+1][c], ...
      float4 q;                                 // pair-interleave: (k,c),(k+1,c),(k,c+1),(k+1,c+1)
      q.x = g0.x * sg; q.y = g1.x * sg; q.z = g0.y * sg; q.w = g1.y * sg;
      *(float4*)(lb + plane * (KC * 64) + kp * 128 + cp * 4) = q;
    }
    __syncthreads();

    if (kc + KC < NPX) {                        // gfx1250 global_prefetch of next A panel
      __builtin_prefetch(arp + kc + KC, 0, 3);
      __builtin_prefetch(aip + kc + KC, 0, 3);
    }

    // ---- compute: 4 k-steps of K=4, 16 WMMAs each ----
    #pragma unroll
    for (int kk4 = 0; kk4 < 4; ++kk4) {
      const int kb = kc + kk4 * 4 + 2 * hh;
      v2f a  = *(const v2f*)(arp + kb);
      v2f ai = *(const v2f*)(aip + kb);
      ai *= sgnA;
      v2f ain = -ai;
      const int kp = kk4 * 2 + hh;              // local k-pair index
      const float* lr = lb + kp * 128 + rr * 2;
      const float* li = lb + (KC * 64) + kp * 128 + rr * 2;
      #pragma unroll
      for (int j = 0; j < 4; ++j) {
        v2f b  = *(const v2f*)(lr + j * 32);    // one ds_load_b64
        v2f bi = *(const v2f*)(li + j * 32);
        // Cr += Ar*Br ; Cr += (-Ai)*Bi ; Ci += Ar*Bi ; Ci += Ai*Br
        cr[j] = __builtin_amdgcn_wmma_f32_16x16x4_f32(false, a,   false, b,  (short)0, cr[j], false, false);
        cr[j] = __builtin_amdgcn_wmma_f32_16x16x4_f32(false, ain, false, bi, (short)0, cr[j], false, false);
        ci[j] = __builtin_amdgcn_wmma_f32_16x16x4_f32(false, a,   false, bi, (short)0, ci[j], false, false);
        ci[j] = __builtin_amdgcn_wmma_f32_16x16x4_f32(false, ai,  false, b,  (short)0, ci[j], false, false);
      }
    }
  }

  #pragma unroll
  for (int j = 0; j < 4; ++j) {
    #pragma unroll
    for (int q = 0; q < 8; ++q) {               // C/D layout: M = q + 8*hh, N col = rr
      const long o = (long)(m0 + q + 8 * hh) * NPX + n0 + j * 16 + rr;
      Cr[o] = cr[j][q];
      Ci[o] = ci[j][q];
    }
  }
}

// ---------------- elementwise kernels ----------------
__global__ __launch_bounds__(256) void k_setf(float* p, float v, long n) {
  long i = (long)blockIdx.x * blockDim.x + threadIdx.x;
  if (i < n) p[i] = v;
}

// out[b,c,h,w] = mask[b,0,h,w] * in[b,c,h,w]
__global__ __launch_bounds__(256) void k_maskmul(float* __restrict__ out,
                                                 const float* __restrict__ in,
                                                 const float* __restrict__ mask) {
  long i = (long)blockIdx.x * blockDim.x + threadIdx.x;
  if (i >= C2) return;
  long b = i / (2 * HW);
  long hw = i % HW;
  out[i] = mask[b * HW + hw] * in[i];
}

// g (B,4,H,W): [dx_re, dx_im, dy_re, dy_im] of x (B,2,H,W)
__global__ __launch_bounds__(256) void k_grad(float* __restrict__ g,
                                              const float* __restrict__ x) {
  long i = (long)blockIdx.x * blockDim.x + threadIdx.x;
  if (i >= C4) return;
  long b = i / (4 * HW);
  long rem = i % (4 * HW);
  int d = (int)(rem / (2 * HW));
  rem %= 2 * HW;
  int ch = (int)(rem / HW);
  long hw = rem % HW;
  int h = (int)(hw / NPX), w = (int)(hw % NPX);
  long s = b * 2 * HW + (long)ch * HW + hw;
  float v0 = x[s], out;
  if (d == 0) out = (w < NPX - 1) ? x[s + 1] - v0 : 0.0f;
  else        out = (h < NPX - 1) ? x[s + NPX] - v0 : 0.0f;
  g[i] = out;
}

// out = base + scale * div(pA - pB)   (pB/base may be null)
__global__ __launch_bounds__(256) void k_div(float* __restrict__ out,
                                             const float* __restrict__ pA,
                                             const float* __restrict__ pB,
                                             const float* __restrict__ base,
                                             float scale) {
  long i = (long)blockIdx.x * blockDim.x + threadIdx.x;
  if (i >= C2) return;
  long b = i / (2 * HW);
  long rem = i % (2 * HW);
  int ch = (int)(rem / HW);
  long hw = rem % HW;
  int h = (int)(hw / NPX), w = (int)(hw % NPX);

  long bx = b * 4 * HW + (long)ch * HW + (long)h * NPX;            // dx plane
  long by = b * 4 * HW + 2 * HW + (long)ch * HW;                   // dy plane
  auto PX = [&](int ww) { long id = bx + ww; float t = pA[id]; if (pB) t -= pB[id]; return t; };
  auto PY = [&](int hh) { long id = by + (long)hh * NPX + w; float t = pA[id]; if (pB) t -= pB[id]; return t; };

  float dvx = (w == 0) ? PX(0) : ((w == NPX - 1) ? -PX(NPX - 2) : PX(w) - PX(w - 1));
  float dvy = (h == 0) ? PY(0) : ((h == NPX - 1) ? -PY(NPX - 2) : PY(h) - PY(h - 1));
  float r = scale * (dvx + dvy);
  if (base) r += base[i];
  out[i] = r;
}

// unew = soft(g + v, thr)
__global__ __launch_bounds__(256) void k_soft(float* __restrict__ unew,
                                              const float* __restrict__ g,
                                              const float* __restrict__ v,
                                              float thr) {
  long i = (long)blockIdx.x * blockDim.x + threadIdx.x;
  if (i >= C4) return;
  float t = g[i] + v[i];
  float a = fabsf(t) - thr;
  a = a > 0.0f ? a : 0.0f;
  unew[i] = copysignf(a, t);
}

// gated ADMM update: u += gate*(unew-u); v += gate*(g-unew)
__global__ __launch_bounds__(256) void k_admm_uv(float* __restrict__ u,
                                                 float* __restrict__ v,
                                                 const float* __restrict__ g,
                                                 const float* __restrict__ unew,
                                                 const float* __restrict__ SC) {
  long i = (long)blockIdx.x * blockDim.x + threadIdx.x;
  if (i >= C4) return;
  float gate = (SC[S_ADONE] == 0.0f) ? 1.0f : 0.0f;
  u[i] += gate * (unew[i] - u[i]);
  v[i] += gate * (g[i] - unew[i]);
}

// r = b - Ap ; p = r
__global__ __launch_bounds__(256) void k_residual(float* __restrict__ r,
                                                  float* __restrict__ p,
                                                  const float* __restrict__ bb,
                                                  const float* __restrict__ Ap) {
  long i = (long)blockIdx.x * blockDim.x + threadIdx.x;
  if (i >= C2) return;
  float t = bb[i] - Ap[i];
  r[i] = t;
  p[i] = t;
}

// gated: x += a*p ; r -= a*Ap
__global__ __launch_bounds__(256) void k_axpy(float* __restrict__ x,
                                              float* __restrict__ r,
                                              const float* __restrict__ p,
                                              const float* __restrict__ Ap,
                                              const float* __restrict__ SC) {
  long i = (long)blockIdx.x * blockDim.x + threadIdx.x;
  if (i >= C2) return;
  float gate = (SC[S_CGDONE] == 0.0f) ? 1.0f : 0.0f;
  float a = gate * SC[S_ALPHA];
  x[i] += a * p[i];
  r[i] -= a * Ap[i];
}

// gated: p = r + beta*p
__global__ __launch_bounds__(256) void k_pupd(float* __restrict__ p,
                                              const float* __restrict__ r,
                                              const float* __restrict__ SC) {
  long i = (long)blockIdx.x * blockDim.x + threadIdx.x;
  if (i >= C2) return;
  if (SC[S_UPD] != 0.0f) p[i] = r[i] + SC[S_BETA] * p[i];
}

// ---------------- deterministic 2-stage reductions ----------------
#define RBLK 512
__global__ __launch_bounds__(256) void k_dot(const float* __restrict__ a,
                                             const float* __restrict__ b,
                                             float* __restrict__ part, long n) {
  __shared__ float sm[256];
  float s = 0.0f;
  for (long i = (long)blockIdx.x * blockDim.x + threadIdx.x; i < n;
       i += (long)gridDim.x * blockDim.x)
    s += a[i] * b[i];
  sm[threadIdx.x] = s;
  __syncthreads();
  for (int off = 128; off > 0; off >>= 1) {
    if ((int)threadIdx.x < off) sm[threadIdx.x] += sm[threadIdx.x + off];
    __syncthreads();
  }
  if (threadIdx.x == 0) part[blockIdx.x] = sm[0];
}

__global__ __launch_bounds__(256) void k_dotdiff(const float* __restrict__ a,
                                                 const float* __restrict__ b,
                                                 float* __restrict__ part, long n) {
  __shared__ float sm[256];
  float s = 0.0f;
  for (long i = (long)blockIdx.x * blockDim.x + threadIdx.x; i < n;
       i += (long)gridDim.x * blockDim.x) {
    float d = a[i] - b[i];
    s += d * d;
  }
  sm[threadIdx.x] = s;
  __syncthreads();
  for (int off = 128; off > 0; off >>= 1) {
    if ((int)threadIdx.x < off) sm[threadIdx.x] += sm[threadIdx.x + off];
    __syncthreads();
  }
  if (threadIdx.x == 0) part[blockIdx.x] = sm[0];
}

__global__ __launch_bounds__(512) void k_reduce512(const float* __restrict__ part,
                                                   float* __restrict__ dst) {
  __shared__ float sm[RBLK];
  sm[threadIdx.x] = part[threadIdx.x];
  __syncthreads();
  for (int off = RBLK / 2; off > 0; off >>= 1) {
    if ((int)threadIdx.x < off) sm[threadIdx.x] += sm[threadIdx.x + off];
    __syncthreads();
  }
  if (threadIdx.x == 0) dst[0] = sm[0];
}

// ---------------- scalar state machines ----------------
__global__ void k_cg_init(float* SC) {
  SC[S_RSOLD] = SC[S_ACC0];
  SC[S_CGDONE] = 0.0f;
}
__global__ void k_cg_alpha(float* SC) {
  SC[S_ALPHA] = SC[S_RSOLD] / (SC[S_ACC0] + 1e-12f);
}
__global__ void k_cg_beta(float* SC) {
  float rsnew = SC[S_ACC1];
  float stop = (rsnew < 1e-10f) ? 1.0f : 0.0f;
  float done = SC[S_CGDONE];
  float upd = (done != 0.0f || stop != 0.0f) ? 0.0f : 1.0f;
  SC[S_BETA] = rsnew / (SC[S_RSOLD] + 1e-30f);
  SC[S_UPD] = upd;
  if (upd != 0.0f) SC[S_RSOLD] = rsnew;
  SC[S_CGDONE] = (done != 0.0f || stop != 0.0f) ? 1.0f : 0.0f;
}
__global__ void k_admm_stats(float* SC) {
  float pres = sqrtf(SC[S_ACC0]);
  float dres = sqrtf(SC[S_ACC1]);
  if (SC[S_ADONE] == 0.0f) {
    SC[S_PRES] = pres;
    SC[S_DRES] = dres;
    SC[S_COUNT] += 1.0f;
    if (pres < 1e-4f && dres < 1e-4f) SC[S_ADONE] = 1.0f;
  }
}

__global__ __launch_bounds__(256) void k_output(float* __restrict__ out,
                                                const float* __restrict__ x,
                                                const float* __restrict__ SC) {
  long i = (long)blockIdx.x * blockDim.x + threadIdx.x;
  if (i < C2) out[i] = x[i];
  else if (i == C2)     out[i] = SC[S_COUNT];
  else if (i == C2 + 1) out[i] = SC[S_PRES];
  else if (i == C2 + 2) out[i] = SC[S_DRES];
}

// ---------------- host orchestration ----------------
extern "C" void kernel_launch(void* const* d_in, const int* in_sizes, int n_in,
                              void* d_out, int out_size, void* d_ws, size_t ws_size,
                              hipStream_t stream) {
  const float* y    = (const float*)d_in[0];  // (4,2,320,320)
  const float* mask = (const float*)d_in[1];  // (4,1,320,320)
  float* W = (float*)d_ws;

  long off = 0;
  float* Fr = W + off; off += NN2;
  float* Fi = W + off; off += NN2;
  float* SC = W + off; off += 16;
  float* PT = W + off; off += RBLK;
  float* x     = W + off; off += C2;
  float* bdata = W + off; off += C2;
  float* bb    = W + off; off += C2;
  float* r     = W + off; off += C2;
  float* p     = W + off; off += C2;
  float* Ap    = W + off; off += C2;
  float* t1    = W + off; off += C2;
  float* t2    = W + off; off += C2;
  float* u     = W + off; off += C4;
  float* v     = W + off; off += C4;
  float* g     = W + off; off += C4;
  float* unew  = W + off; off += C4;

  const dim3 GEMM_G(5, 5, NB), GEMM_B(32, 4);   // 64x64 macro-tiles
  const int E2 = (int)((C2 + 255) / 256);
  const int E4 = (int)((C4 + 255) / 256);

  auto CGEMM = [&](const float* Ar, const float* Ai, long sA, float gA,
                   const float* Br, const float* Bi, long sB, float gB,
                   float* Cr, float* Ci, long sC) {
    k_cgemm<<<GEMM_G, GEMM_B, 0, stream>>>(Ar, Ai, sA, gA, Br, Bi, sB, gB, Cr, Ci, sC);
  };
  // out = (sign>0 ? fft2c : ifft2c)(in); in/out are (B,2,H,W); F is symmetric.
  auto FFT2 = [&](const float* in, float* out, float sign) {
    CGEMM(Fr, Fi, 0, sign, in, in + HW, 2 * HW, 1.0f, t1, t1 + HW, 2 * HW);   // T = F*X
    CGEMM(t1, t1 + HW, 2 * HW, 1.0f, Fr, Fi, 0, sign, out, out + HW, 2 * HW); // Y = T*F
  };
  auto DOT = [&](const float* a, const float* b, float* dst, long n) {
    k_dot<<<RBLK, 256, 0, stream>>>(a, b, PT, n);
    k_reduce512<<<1, RBLK, 0, stream>>>(PT, dst);
  };
  auto APPLY_A = [&](const float* z) {  // Ap = ifft2c(mask*fft2c(z)) - RHO*div(grad(z))
    FFT2(z, t2, +1.0f);
    k_maskmul<<<E2, 256, 0, stream>>>(t2, t2, mask);
    FFT2(t2, Ap, -1.0f);
    k_grad<<<E4, 256, 0, stream>>>(g, z);
    k_div<<<E2, 256, 0, stream>>>(Ap, g, nullptr, Ap, -RHO_);
  };
  auto CG = [&]() {
    APPLY_A(x);
    k_residual<<<E2, 256, 0, stream>>>(r, p, bb, Ap);
    DOT(r, r, SC + S_ACC0, C2);
    k_cg_init<<<1, 1, 0, stream>>>(SC);
    for (int it = 0; it < 10; ++it) {
      APPLY_A(p);
      DOT(p, Ap, SC + S_ACC0, C2);
      k_cg_alpha<<<1, 1, 0, stream>>>(SC);
      k_axpy<<<E2, 256, 0, stream>>>(x, r, p, Ap, SC);
      DOT(r, r, SC + S_ACC1, C2);
      k_cg_beta<<<1, 1, 0, stream>>>(SC);
      k_pupd<<<E2, 256, 0, stream>>>(p, r, SC);
    }
  };

  // ---- init ----
  k_build_dft<<<(int)((NN2 + 255) / 256), 256, 0, stream>>>(Fr, Fi);
  k_setf<<<1, 16, 0, stream>>>(SC, 0.0f, 16);
  FFT2(y, x, -1.0f);                                  // x0 = ifft2c(y)
  k_grad<<<E4, 256, 0, stream>>>(u, x);               // u0 = grad(x0)
  k_setf<<<E4, 256, 0, stream>>>(v, 0.0f, C4);        // v0 = 0
  k_maskmul<<<E2, 256, 0, stream>>>(t2, y, mask);
  FFT2(t2, bdata, -1.0f);                             // b_data = ifft2c(mask*y)

  // ---- ADMM ----
  const float thresh = LAMBDA_TV_ / RHO_;
  for (int it = 0; it < 10; ++it) {
    k_div<<<E2, 256, 0, stream>>>(bb, u, v, bdata, RHO_);   // b = bdata + rho*div(u-v)
    CG();                                                    // x <- cg solve
    k_grad<<<E4, 256, 0, stream>>>(g, x);
    k_soft<<<E4, 256, 0, stream>>>(unew, g, v, thresh);
    k_dotdiff<<<RBLK, 256, 0, stream>>>(g, unew, PT, C4);    // pres^2
    k_reduce512<<<1, RBLK, 0, stream>>>(PT, SC + S_ACC0);
    k_div<<<E2, 256, 0, stream>>>(t1, unew, u, nullptr, RHO_); // rho*div(un-u)
    DOT(t1, t1, SC + S_ACC1, C2);                            // dres^2
    k_admm_uv<<<E4, 256, 0, stream>>>(u, v, g, unew, SC);
    k_admm_stats<<<1, 1, 0, stream>>>(SC);
  }

  k_output<<<(int)((C2 + 3 + 255) / 256), 256, 0, stream>>>((float*)d_out, x, SC);
}